// BrainGNNEncoder_16475494547815
// MI455X (gfx1250) — compile-verified
//
#include <hip/hip_runtime.h>
#include <hip/hip_bf16.h>

typedef __attribute__((ext_vector_type(2))) float v2f;
typedef __attribute__((ext_vector_type(8))) float v8f;

#define HC  64      // hidden channels
#define TPB 256
#define BN_EPS 1e-5f

// ---------------- degree / normalization ----------------

__global__ void k_init_deg(float* __restrict__ deg, int n) {
  int i = blockIdx.x * blockDim.x + threadIdx.x;
  if (i < n) deg[i] = 1.0f;                       // self-loop weight
}

__global__ void k_deg_scatter(const int* __restrict__ dst, const float* __restrict__ w,
                              float* __restrict__ deg, int e) {
  int i = blockIdx.x * blockDim.x + threadIdx.x;
  if (i < e) atomicAdd(&deg[dst[i]], fabsf(w[i]));
}

__global__ void k_deg_to_dinv(float* __restrict__ deg, int n) {
  int i = blockIdx.x * blockDim.x + threadIdx.x;
  if (i < n) {
    float d = deg[i];
    deg[i] = (d > 0.0f) ? rsqrtf(d) : 0.0f;
  }
}

__global__ void k_edge_norm(const int* __restrict__ src, const int* __restrict__ dst,
                            const float* __restrict__ w, const float* __restrict__ dinv,
                            float* __restrict__ norm, int e) {
  int i = blockIdx.x * blockDim.x + threadIdx.x;
  if (i < e) norm[i] = dinv[src[i]] * fabsf(w[i]) * dinv[dst[i]];
}

// ---------------- dense GEMM: Y[N,64] = X[N,K] @ W[K,64] (fp32 WMMA) ----------------
// One wave computes a 16-row x 64-col output block with 4 accumulators,
// K-loop in steps of 4 using V_WMMA_F32_16X16X4_F32. EXEC stays all-ones
// for every wave that reaches the WMMA loop (tile exit is wave-uniform).

__global__ void k_gemm_wmma(const float* __restrict__ X, const float* __restrict__ Wm,
                            float* __restrict__ Y, int nrows, int K) {
  int gwave = (int)((blockIdx.x * (unsigned)blockDim.x + threadIdx.x) >> 5);
  int lane  = threadIdx.x & 31;
  int row0  = gwave * 16;
  if (row0 >= nrows) return;                      // wave-uniform exit

  int m     = lane & 15;                          // row (A) / col (B,C) within tile
  int khalf = (lane >> 4) * 2;                    // 0 or 2: K-pair this lane owns

  int arow = row0 + m;
  if (arow >= nrows) arow = nrows - 1;            // clamp loads for ragged tail
  const float* xrow = X + (size_t)arow * (size_t)K;

  v8f acc0 = {}, acc1 = {}, acc2 = {}, acc3 = {};
  for (int k = 0; k < K; k += 4) {
    v2f a;
    a.x = xrow[k + khalf];
    a.y = xrow[k + khalf + 1];
    const float* w0 = Wm + (size_t)(k + khalf)     * HC;
    const float* w1 = Wm + (size_t)(k + khalf + 1) * HC;
    v2f b0, b1, b2, b3;
    b0.x = w0[m];      b0.y = w1[m];
    b1.x = w0[16 + m]; b1.y = w1[16 + m];
    b2.x = w0[32 + m]; b2.y = w1[32 + m];
    b3.x = w0[48 + m]; b3.y = w1[48 + m];
    acc0 = __builtin_amdgcn_wmma_f32_16x16x4_f32(false, a, false, b0, (short)0, acc0, false, false);
    acc1 = __builtin_amdgcn_wmma_f32_16x16x4_f32(false, a, false, b1, (short)0, acc1, false, false);
    acc2 = __builtin_amdgcn_wmma_f32_16x16x4_f32(false, a, false, b2, (short)0, acc2, false, false);
    acc3 = __builtin_amdgcn_wmma_f32_16x16x4_f32(false, a, false, b3, (short)0, acc3, false, false);
  }

  int col   = lane & 15;
  int rbase = (lane >> 4) * 8;                    // rows 0..7 or 8..15 of the tile
#pragma unroll
  for (int i = 0; i < 8; ++i) {
    int r = row0 + rbase + i;
    if (r < nrows) {
      float* yr = Y + (size_t)r * HC;
      yr[col]      = acc0[i];
      yr[16 + col] = acc1[i];
      yr[32 + col] = acc2[i];
      yr[48 + col] = acc3[i];
    }
  }
}

// ---------------- aggregation ----------------
// out[i,f] = bias[f] + dinv[i]^2 * h[i,f]   (self-loop message + bias)

__global__ void k_agg_init(const float* __restrict__ h, const float* __restrict__ dinv,
                           const float* __restrict__ bias, float* __restrict__ out,
                           long total) {
  long t = (long)blockIdx.x * blockDim.x + threadIdx.x;
  if (t < total) {
    int node = (int)(t >> 6);
    int f    = (int)(t & 63);
    float d  = dinv[node];
    out[t] = bias[f] + d * d * h[t];
  }
}

// out[dst[e], f] += norm[e] * h[src[e], f]  -- 64 threads per edge, coalesced rows

__global__ void k_agg_edges(const int* __restrict__ src, const int* __restrict__ dst,
                            const float* __restrict__ norm, const float* __restrict__ h,
                            float* __restrict__ out, int e) {
  long t = (long)blockIdx.x * blockDim.x + threadIdx.x;
  int ei = (int)(t >> 6);
  int f  = (int)(t & 63);
  if (ei < e) {
    float v = norm[ei] * h[(size_t)src[ei] * HC + f];
    atomicAdd(&out[(size_t)dst[ei] * HC + f], v);
  }
}

// BN (eval, folded to affine) + ReLU, in place

__global__ void k_bn_relu(float* __restrict__ h, const float* __restrict__ g,
                          const float* __restrict__ bt, const float* __restrict__ rm,
                          const float* __restrict__ rv, long total) {
  long t = (long)blockIdx.x * blockDim.x + threadIdx.x;
  if (t < total) {
    int f = (int)(t & 63);
    float sc = g[f] * rsqrtf(rv[f] + BN_EPS);
    float v  = (h[t] - rm[f]) * sc + bt[f];
    h[t] = fmaxf(v, 0.0f);
  }
}

// ---------------- pooling ----------------

__global__ void k_pool_zero(float* __restrict__ p, int total) {
  int i = blockIdx.x * blockDim.x + threadIdx.x;
  if (i < total) p[i] = 0.0f;
}

__global__ void k_pool_scatter(const float* __restrict__ h, const int* __restrict__ batch,
                               float* __restrict__ pool, float* __restrict__ cnt, long total) {
  long t = (long)blockIdx.x * blockDim.x + threadIdx.x;
  if (t < total) {
    int node = (int)(t >> 6);
    int f    = (int)(t & 63);
    int b    = batch[node];
    atomicAdd(&pool[(size_t)b * HC + f], h[t]);
    if (f == 0) atomicAdd(&cnt[b], 1.0f);
  }
}

__global__ void k_pool_final(const float* __restrict__ pool, const float* __restrict__ cnt,
                             float* __restrict__ out, int g64) {
  int t = blockIdx.x * blockDim.x + threadIdx.x;
  if (t < g64) {
    int g = t >> 6;
    int f = t & 63;
    float s = pool[t];
    float c = fmaxf(cnt[g], 1.0f);
    out[(size_t)g * (2 * HC) + f]       = s / c;   // mean
    out[(size_t)g * (2 * HC) + HC + f]  = s;       // sum
  }
}

// ---------------- launch ----------------

static inline size_t align256(size_t v) { return (v + 255) & ~(size_t)255; }
static inline int nblk(long total) { return (int)((total + TPB - 1) / TPB); }

extern "C" void kernel_launch(void* const* d_in, const int* in_sizes, int n_in,
                              void* d_out, int out_size, void* d_ws, size_t ws_size,
                              hipStream_t stream) {
  const float* x     = (const float*)d_in[0];
  const int*   ei    = (const int*)d_in[1];
  const float* ew    = (const float*)d_in[2];
  const int*   batch = (const int*)d_in[3];

  const int E  = in_sizes[2];
  const int n  = in_sizes[3];
  const int IN = in_sizes[0] / n;
  const int G  = out_size / (2 * HC);

  const int* src = ei;       // edge_index[0, :]
  const int* dst = ei + E;   // edge_index[1, :]

  const float *W[3], *bia[3], *gg[3], *bt[3], *rm[3], *rv[3];
  for (int l = 0; l < 3; ++l) {
    int base = 4 + 6 * l;
    W[l]   = (const float*)d_in[base + 0];
    bia[l] = (const float*)d_in[base + 1];
    gg[l]  = (const float*)d_in[base + 2];
    bt[l]  = (const float*)d_in[base + 3];
    rm[l]  = (const float*)d_in[base + 4];
    rv[l]  = (const float*)d_in[base + 5];
  }

  // workspace partition
  char* p = (char*)d_ws;
  float* dinv = (float*)p; p += align256((size_t)n * 4);
  float* norm = (float*)p; p += align256((size_t)E * 4);
  float* hA   = (float*)p; p += align256((size_t)n * HC * 4);
  float* hB   = (float*)p; p += align256((size_t)n * HC * 4);
  float* pool = (float*)p; p += align256((size_t)G * HC * 4);
  float* cnt  = (float*)p;

  const long nh   = (long)n * HC;          // node-feature elements
  const long eh   = (long)E * HC;          // edge-feature work items
  const int  ntiles = (n + 15) / 16;
  const int  gemm_blocks = (ntiles + (TPB / 32) - 1) / (TPB / 32);

  // --- normalization (done once) ---
  hipLaunchKernelGGL(k_init_deg,    dim3(nblk(n)), dim3(TPB), 0, stream, dinv, n);
  hipLaunchKernelGGL(k_deg_scatter, dim3(nblk(E)), dim3(TPB), 0, stream, dst, ew, dinv, E);
  hipLaunchKernelGGL(k_deg_to_dinv, dim3(nblk(n)), dim3(TPB), 0, stream, dinv, n);
  hipLaunchKernelGGL(k_edge_norm,   dim3(nblk(E)), dim3(TPB), 0, stream, src, dst, ew, dinv, norm, E);

  // --- 3 GCN layers: GEMM (WMMA f32) -> scatter-agg -> BN+ReLU ---
  const float* hin = x;
  int K = IN;
  for (int l = 0; l < 3; ++l) {
    hipLaunchKernelGGL(k_gemm_wmma, dim3(gemm_blocks), dim3(TPB), 0, stream,
                       hin, W[l], hA, n, K);
    hipLaunchKernelGGL(k_agg_init,  dim3(nblk(nh)), dim3(TPB), 0, stream,
                       hA, dinv, bia[l], hB, nh);
    hipLaunchKernelGGL(k_agg_edges, dim3(nblk(eh)), dim3(TPB), 0, stream,
                       src, dst, norm, hA, hB, E);
    hipLaunchKernelGGL(k_bn_relu,   dim3(nblk(nh)), dim3(TPB), 0, stream,
                       hB, gg[l], bt[l], rm[l], rv[l], nh);
    hin = hB;   // next layer reads hB, writes hA
    K = HC;
  }

  // --- pooling: [mean || sum] per graph ---
  int ptotal = G * HC + G;  // pool + cnt are contiguous
  hipLaunchKernelGGL(k_pool_zero,    dim3(nblk(ptotal)), dim3(TPB), 0, stream, pool, ptotal);
  hipLaunchKernelGGL(k_pool_scatter, dim3(nblk(nh)),     dim3(TPB), 0, stream, hB, batch, pool, cnt, nh);
  hipLaunchKernelGGL(k_pool_final,   dim3(nblk(G * HC)), dim3(TPB), 0, stream, pool, cnt, (float*)d_out, G * HC);

  (void)n_in; (void)ws_size;
}